// GaussianModel_36670430773888
// MI455X (gfx1250) — compile-verified
//
#include <hip/hip_runtime.h>

#define BLOCK 256

// Native clang vectors (HIP's float4 class is rejected by the nontemporal /
// async builtins).
typedef float v4f __attribute__((ext_vector_type(4)));
typedef int   v4i __attribute__((ext_vector_type(4)));
typedef __attribute__((address_space(1))) v4i gv4i;  // global int4
typedef __attribute__((address_space(3))) v4i lv4i;  // LDS int4

// CDNA5 async LDS->global store path, if this toolchain exposes the builtins.
#if defined(__has_builtin)
#  if __has_builtin(__builtin_amdgcn_global_store_async_from_lds_b128) && \
      __has_builtin(__builtin_amdgcn_s_wait_asynccnt)
#    define USE_ASYNC_STORE 1
#  endif
#endif

__global__ __launch_bounds__(BLOCK) void quat_scale_cov_kernel(
    const float* __restrict__ rot,   // (N,4) w,x,y,z
    const float* __restrict__ scl,   // (N,3)
    float* __restrict__ out,         // (N,3,3)
    int n)
{
    // Per-block staging buffer: 256 points * 9 floats = 9216 B, 16B aligned.
    __shared__ __align__(16) float lbuf[BLOCK * 9];

    const int tid        = threadIdx.x;
    const long long base = (long long)blockIdx.x * BLOCK;
    const long long idx  = base + tid;
    const bool active    = idx < (long long)n;

    float w = 0.f, x = 0.f, y = 0.f, z = 0.f;
    float sx = 0.f, sy = 0.f, sz = 0.f;
    if (active) {
        // Coalesced 16B load per lane (global_load_b128), streamed-once -> NT hint.
        const v4f q = __builtin_nontemporal_load(((const v4f*)rot) + idx);
        w = q.x; x = q.y; y = q.z; z = q.w;
        const float* sp = scl + idx * 3;
        sx = __builtin_nontemporal_load(sp + 0);
        sy = __builtin_nontemporal_load(sp + 1);
        sz = __builtin_nontemporal_load(sp + 2);
    }

    // R entries are quadratic in q/|q|, so only 1/|q|^2 is needed (no sqrt).
    const float n2 = w*w + x*x + y*y + z*z;
    const float t  = 2.0f * __builtin_amdgcn_rcpf(n2);   // v_rcp_f32

    const float wx = w*x, wy = w*y, wz = w*z;
    const float xx = x*x, xy = x*y, xz = x*z;
    const float yy = y*y, yz = y*z, zz = z*z;

    const float a00 = 1.0f - t*(yy + zz);
    const float a01 = t*(xy - wz);
    const float a02 = t*(xz + wy);
    const float a10 = t*(xy + wz);
    const float a11 = 1.0f - t*(xx + zz);
    const float a12 = t*(yz - wx);
    const float a20 = t*(xz - wy);
    const float a21 = t*(yz + wx);
    const float a22 = 1.0f - t*(xx + yy);

    const float sx2 = sx*sx, sy2 = sy*sy, sz2 = sz*sz;

    // cov[i][k] = sum_j R[i][j]*R[k][j]*s[j]^2  (symmetric, 6 unique values)
    const float c00 = a00*a00*sx2 + a01*a01*sy2 + a02*a02*sz2;
    const float c01 = a00*a10*sx2 + a01*a11*sy2 + a02*a12*sz2;
    const float c02 = a00*a20*sx2 + a01*a21*sy2 + a02*a22*sz2;
    const float c11 = a10*a10*sx2 + a11*a11*sy2 + a12*a12*sz2;
    const float c12 = a10*a20*sx2 + a11*a21*sy2 + a12*a22*sz2;
    const float c22 = a20*a20*sx2 + a21*a21*sy2 + a22*a22*sz2;

    // Stage into LDS in global output order. Stride 9 is coprime with the 64
    // banks -> conflict-free writes.
    float* l = &lbuf[tid * 9];
    l[0] = c00; l[1] = c01; l[2] = c02;
    l[3] = c01; l[4] = c11; l[5] = c12;
    l[6] = c02; l[7] = c12; l[8] = c22;

    __syncthreads();

    // Drain the block's 9216 B contiguously: 576 x 16B transfers.
    float* outb = out + base * 9;                 // 16B aligned (9216 * blockIdx)
    const long long remaining = (long long)n - base;

    if (remaining >= BLOCK) {
#if defined(USE_ASYNC_STORE)
        for (int j4 = tid; j4 < (BLOCK * 9) / 4; j4 += BLOCK) {
            __builtin_amdgcn_global_store_async_from_lds_b128(
                (gv4i*)(outb + 4 * j4), (lv4i*)(lbuf + 4 * j4), 0, 0);
        }
        __builtin_amdgcn_s_wait_asynccnt(0);
#else
        v4f* __restrict__ o4       = (v4f*)outb;
        const v4f* __restrict__ l4 = (const v4f*)lbuf;
        for (int j4 = tid; j4 < (BLOCK * 9) / 4; j4 += BLOCK) {
            __builtin_nontemporal_store(l4[j4], o4 + j4);
        }
#endif
    } else if (remaining > 0) {
        const int tot = (int)remaining * 9;
        for (int j = tid; j < tot; j += BLOCK) outb[j] = lbuf[j];
    }
}

extern "C" void kernel_launch(void* const* d_in, const int* in_sizes, int n_in,
                              void* d_out, int out_size, void* d_ws, size_t ws_size,
                              hipStream_t stream) {
    const float* rot = (const float*)d_in[0];   // (N,4) float32
    const float* scl = (const float*)d_in[1];   // (N,3) float32
    float* out = (float*)d_out;                 // (N,3,3) float32
    const int n = in_sizes[0] / 4;
    const int blocks = (n + BLOCK - 1) / BLOCK; // 31250 for N=8e6
    quat_scale_cov_kernel<<<blocks, BLOCK, 0, stream>>>(rot, scl, out, n);
}